// DrugGNN_89541478187306
// MI455X (gfx1250) — compile-verified
//
#include <hip/hip_runtime.h>
#include <hip/hip_bf16.h>

typedef __attribute__((ext_vector_type(2))) float v2f;
typedef __attribute__((ext_vector_type(8))) float v8f;

#define N_NODES 100000
#define E_EDGES 1000000
#define FIN 64
#define HID 128
#define NCLS 32
#define NGRAPH 1024
#define BN_EPS 1e-5f

// ---------------- utility kernels ----------------

__global__ void fill_kernel(float* __restrict__ p, int n, float v) {
    int i = blockIdx.x * blockDim.x + threadIdx.x;
    if (i < n) p[i] = v;
}

__global__ void deg_count_kernel(const int* __restrict__ ei, float* __restrict__ deg) {
    int i = blockIdx.x * blockDim.x + threadIdx.x;
    if (i < E_EDGES) atomicAdd(&deg[ei[E_EDGES + i]], 1.0f);  // dst row of edge_index
}

__global__ void rsqrt_kernel(float* __restrict__ p, int n) {
    int i = blockIdx.x * blockDim.x + threadIdx.x;
    if (i < n) p[i] = rsqrtf(p[i]);
}

// ---------------- GEMM1: h = x @ W_gcn  (fp32 WMMA 16x16x4) ----------------
// A frag (16x4 f32): a[v] @lane l = A[l&15][2*(l>>4)+v]
// B frag (4x16 f32): b[v] @lane l = B[2*(l>>4)+v][l&15]
// C/D (16x16 f32):   d[v] @lane l = D[v+8*(l>>4)][l&15]

__global__ __launch_bounds__(128) void gemm1_wmma(const float* __restrict__ x,
                                                  const float* __restrict__ Wg,
                                                  float* __restrict__ h) {
    __shared__ float wlds[FIN * HID];  // 32 KB
    const int tid = threadIdx.x;
    for (int i = tid; i < FIN * HID; i += 128) wlds[i] = Wg[i];
    __syncthreads();

    const int lane  = tid & 31;
    const int wave  = tid >> 5;
    const int tile  = blockIdx.x * 4 + wave;      // 16-row tile index
    if (tile >= N_NODES / 16) return;             // wave-uniform exit (EXEC stays all-1)
    const int m0    = tile * 16;
    const int mrow  = lane & 15;
    const int khalf = (lane >> 4) * 2;
    const int hihalf = 8 * (lane >> 4);

    // Preload all A fragments for this 16-row strip (K = 64 -> 16 steps of 4)
    v2f afrag[16];
#pragma unroll
    for (int kk = 0; kk < 16; ++kk) {
        const float* ap = &x[(size_t)(m0 + mrow) * FIN + kk * 4 + khalf];
        afrag[kk].x = ap[0];
        afrag[kk].y = ap[1];
    }

#pragma unroll
    for (int nt = 0; nt < 8; ++nt) {              // HID = 8 tiles of 16 columns
        const int n0 = nt * 16;
        v8f acc = {};
#pragma unroll
        for (int kk = 0; kk < 16; ++kk) {
            const int kb = kk * 4 + khalf;
            v2f b;
            b.x = wlds[kb * HID + n0 + mrow];
            b.y = wlds[(kb + 1) * HID + n0 + mrow];
            acc = __builtin_amdgcn_wmma_f32_16x16x4_f32(false, afrag[kk], false, b,
                                                        (short)0, acc, false, false);
        }
#pragma unroll
        for (int v = 0; v < 8; ++v)
            h[(size_t)(m0 + v + hihalf) * HID + n0 + mrow] = acc[v];
    }
}

// ---------------- self-loop init: h2[n] = h[n] * dinv[n]^2 ----------------

__global__ void self_init_kernel(const float* __restrict__ h, const float* __restrict__ dinv,
                                 float* __restrict__ h2) {
    int i = blockIdx.x * blockDim.x + threadIdx.x;
    if (i >= N_NODES * HID) return;
    int n = i >> 7;
    float dv = dinv[n];
    h2[i] = h[i] * dv * dv;
}

// ---------------- edge aggregation: h2[dst] += h[src] * dinv[src]*dinv[dst] ----------------

__global__ void edge_agg_kernel(const int* __restrict__ ei, const float* __restrict__ h,
                                const float* __restrict__ dinv, float* __restrict__ h2) {
    int gid = blockIdx.x * blockDim.x + threadIdx.x;   // E * 32 chunks (4 floats each)
    if (gid >= E_EDGES * (HID / 4)) return;
    int e  = gid >> 5;
    int f0 = (gid & 31) * 4;
    int s  = ei[e];
    int d  = ei[E_EDGES + e];
    float coef = dinv[s] * dinv[d];
    const float4 hv = *(const float4*)&h[(size_t)s * HID + f0];
    float* o = &h2[(size_t)d * HID + f0];
    atomicAdd(o + 0, hv.x * coef);
    atomicAdd(o + 1, hv.y * coef);
    atomicAdd(o + 2, hv.z * coef);
    atomicAdd(o + 3, hv.w * coef);
}

// ---------------- BN1 column stats ----------------

__global__ __launch_bounds__(HID) void bn1_stats_kernel(const float* __restrict__ h2,
                                                        float* __restrict__ ssum,
                                                        float* __restrict__ ssq) {
    const int f  = threadIdx.x;
    const int r0 = blockIdx.x * 256;
    int r1 = r0 + 256; if (r1 > N_NODES) r1 = N_NODES;
    float s = 0.f, q = 0.f;
    for (int r = r0; r < r1; ++r) {
        float v = h2[(size_t)r * HID + f];
        s += v; q += v * v;
    }
    atomicAdd(&ssum[f], s);
    atomicAdd(&ssq[f], q);
}

__global__ __launch_bounds__(HID) void bn1_finalize_kernel(const float* __restrict__ ssum,
                                                           const float* __restrict__ ssq,
                                                           const float* __restrict__ w,
                                                           const float* __restrict__ b,
                                                           float* __restrict__ sc,
                                                           float* __restrict__ sh) {
    int f = threadIdx.x;
    float mu  = ssum[f] * (1.0f / N_NODES);
    float var = ssq[f] * (1.0f / N_NODES) - mu * mu;
    float a   = w[f] * rsqrtf(var + BN_EPS);
    sc[f] = a;
    sh[f] = b[f] - mu * a;
}

// ---------------- graph counts + fused BN1/ReLU/pool-sum ----------------

__global__ void cnt_kernel(const int* __restrict__ batch, float* __restrict__ cnt) {
    int i = blockIdx.x * blockDim.x + threadIdx.x;
    if (i < N_NODES) atomicAdd(&cnt[batch[i]], 1.0f);
}

__global__ void pool_kernel(const float* __restrict__ h2, const int* __restrict__ batch,
                            const float* __restrict__ sc, const float* __restrict__ sh,
                            float* __restrict__ psum) {
    int i = blockIdx.x * blockDim.x + threadIdx.x;
    if (i >= N_NODES * HID) return;
    int n = i >> 7, f = i & (HID - 1);
    float v = fmaf(sc[f], h2[i], sh[f]);
    v = fmaxf(v, 0.0f);
    atomicAdd(&psum[(size_t)batch[n] * HID + f], v);
}

__global__ void pooled_div_kernel(const float* __restrict__ psum, const float* __restrict__ cnt,
                                  float* __restrict__ pooled) {
    int i = blockIdx.x * blockDim.x + threadIdx.x;
    if (i >= NGRAPH * HID) return;
    int g = i >> 7;
    pooled[i] = psum[i] / fmaxf(cnt[g], 1.0f);
}

// ---------------- GEMM2: out_pre = pooled @ W_out  (fp32 WMMA) ----------------

__global__ __launch_bounds__(64) void gemm2_wmma(const float* __restrict__ pooled,
                                                 const float* __restrict__ Wo,
                                                 float* __restrict__ outp) {
    __shared__ float wl[HID * NCLS];  // 16 KB
    const int tid = threadIdx.x;
    for (int i = tid; i < HID * NCLS; i += 64) wl[i] = Wo[i];
    __syncthreads();

    const int lane  = tid & 31;
    const int n0    = (tid >> 5) * 16;     // wave 0 -> cols 0..15, wave 1 -> 16..31
    const int m0    = blockIdx.x * 16;
    const int mrow  = lane & 15;
    const int khalf = (lane >> 4) * 2;

    v8f acc = {};
#pragma unroll
    for (int kk = 0; kk < 32; ++kk) {      // K = 128 -> 32 steps of 4
        const int kb = kk * 4 + khalf;
        v2f a, b;
        const float* ap = &pooled[(size_t)(m0 + mrow) * HID + kb];
        a.x = ap[0];
        a.y = ap[1];
        b.x = wl[kb * NCLS + n0 + mrow];
        b.y = wl[(kb + 1) * NCLS + n0 + mrow];
        acc = __builtin_amdgcn_wmma_f32_16x16x4_f32(false, a, false, b,
                                                    (short)0, acc, false, false);
    }
#pragma unroll
    for (int v = 0; v < 8; ++v)
        outp[(size_t)(m0 + v + 8 * (lane >> 4)) * NCLS + n0 + mrow] = acc[v];
}

// ---------------- BN2 ----------------

__global__ __launch_bounds__(NCLS) void bn2_stats_kernel(const float* __restrict__ outp,
                                                         const float* __restrict__ w,
                                                         const float* __restrict__ b,
                                                         float* __restrict__ ab) {
    int f = threadIdx.x;
    float s = 0.f, q = 0.f;
    for (int g = 0; g < NGRAPH; ++g) {
        float v = outp[g * NCLS + f];
        s += v; q += v * v;
    }
    float mu  = s * (1.0f / NGRAPH);
    float var = q * (1.0f / NGRAPH) - mu * mu;
    float a   = w[f] * rsqrtf(var + BN_EPS);
    ab[f] = a;
    ab[NCLS + f] = b[f] - mu * a;
}

__global__ void bn2_apply_kernel(const float* __restrict__ outp, const float* __restrict__ ab,
                                 float* __restrict__ out) {
    int i = blockIdx.x * blockDim.x + threadIdx.x;
    if (i >= NGRAPH * NCLS) return;
    int f = i & (NCLS - 1);
    out[i] = fmaf(ab[f], outp[i], ab[NCLS + f]);
}

// ---------------- host launcher ----------------

extern "C" void kernel_launch(void* const* d_in, const int* in_sizes, int n_in,
                              void* d_out, int out_size, void* d_ws, size_t ws_size,
                              hipStream_t stream) {
    (void)in_sizes; (void)n_in; (void)out_size; (void)ws_size;
    const float* x      = (const float*)d_in[0];
    const int*   ei     = (const int*)d_in[1];     // [2, E]
    const int*   batch  = (const int*)d_in[2];     // [N]
    const float* W_gcn  = (const float*)d_in[3];
    // d_in[4] = b_gcn: constant per-feature bias before training-mode BN => no-op
    const float* bn1_w  = (const float*)d_in[5];
    const float* bn1_b  = (const float*)d_in[6];
    const float* W_out  = (const float*)d_in[7];
    // d_in[8] = b_out: no-op through BN2
    const float* bn2_w  = (const float*)d_in[9];
    const float* bn2_b  = (const float*)d_in[10];
    float* out = (float*)d_out;

    // carve workspace
    char* base = (char*)d_ws;
    size_t off = 0;
    auto alloc = [&](size_t bytes) -> float* {
        float* p = (float*)(base + off);
        off = (off + bytes + 255) & ~(size_t)255;
        return p;
    };
    float* dinv   = alloc((size_t)N_NODES * 4);            // deg -> dinv in place
    float* h      = alloc((size_t)N_NODES * HID * 4);
    float* h2     = alloc((size_t)N_NODES * HID * 4);
    float* ssum   = alloc(HID * 4);
    float* ssq    = alloc(HID * 4);
    float* sc1    = alloc(HID * 4);
    float* sh1    = alloc(HID * 4);
    float* psum   = alloc((size_t)NGRAPH * HID * 4);
    float* cnt    = alloc(NGRAPH * 4);
    float* pooled = alloc((size_t)NGRAPH * HID * 4);
    float* outp   = alloc((size_t)NGRAPH * NCLS * 4);
    float* ab2    = alloc(2 * NCLS * 4);

    // 1. degrees (self loop contributes 1)
    fill_kernel<<<(N_NODES + 255) / 256, 256, 0, stream>>>(dinv, N_NODES, 1.0f);
    deg_count_kernel<<<(E_EDGES + 255) / 256, 256, 0, stream>>>(ei, dinv);
    rsqrt_kernel<<<(N_NODES + 255) / 256, 256, 0, stream>>>(dinv, N_NODES);

    // 2. h = x @ W_gcn  (WMMA)
    gemm1_wmma<<<(N_NODES / 16 + 3) / 4, 128, 0, stream>>>(x, W_gcn, h);

    // 3. aggregate with symmetric norm
    self_init_kernel<<<(N_NODES * HID + 255) / 256, 256, 0, stream>>>(h, dinv, h2);
    edge_agg_kernel<<<(E_EDGES * (HID / 4) + 255) / 256, 256, 0, stream>>>(ei, h, dinv, h2);

    // 4. BN1 stats -> scale/shift
    fill_kernel<<<1, 256, 0, stream>>>(ssum, 2 * HID, 0.0f);   // ssum & ssq contiguous
    bn1_stats_kernel<<<(N_NODES + 255) / 256, HID, 0, stream>>>(h2, ssum, ssq);
    bn1_finalize_kernel<<<1, HID, 0, stream>>>(ssum, ssq, bn1_w, bn1_b, sc1, sh1);

    // 5. fused BN+ReLU + mean pool
    fill_kernel<<<(NGRAPH * HID + 255) / 256, 256, 0, stream>>>(psum, NGRAPH * HID, 0.0f);
    fill_kernel<<<(NGRAPH + 255) / 256, 256, 0, stream>>>(cnt, NGRAPH, 0.0f);
    cnt_kernel<<<(N_NODES + 255) / 256, 256, 0, stream>>>(batch, cnt);
    pool_kernel<<<(N_NODES * HID + 255) / 256, 256, 0, stream>>>(h2, batch, sc1, sh1, psum);
    pooled_div_kernel<<<(NGRAPH * HID + 255) / 256, 256, 0, stream>>>(psum, cnt, pooled);

    // 6. head GEMM (WMMA) + BN2
    gemm2_wmma<<<NGRAPH / 16, 64, 0, stream>>>(pooled, W_out, outp);
    bn2_stats_kernel<<<1, NCLS, 0, stream>>>(outp, bn2_w, bn2_b, ab2);
    bn2_apply_kernel<<<(NGRAPH * NCLS + 255) / 256, 256, 0, stream>>>(outp, ab2, out);
}